// Inv2d_42537356099835
// MI455X (gfx1250) — compile-verified
//
#include <hip/hip_runtime.h>

typedef float v2f __attribute__((ext_vector_type(2)));
typedef float v8f __attribute__((ext_vector_type(8)));

#define Bn   4
#define Cn   128
#define Hn   64
#define Wn   64
#define Gn   8
#define CgN  16
#define Kk   7
#define KKn  49
#define KKG  392
#define CrN  32
#define DILn 2
#define PADn 6
#define TPX  16   // pixels per tile (WMMA N dimension)

// Fused involution: per 16-pixel tile,
//  phase 1: stage x[:,tile] channel-major into LDS
//  phase 2: red = W_reduce @ x_tile        (WMMA f32 16x16x4, 2 M-tiles)
//  phase 3: ker = W_span @ red             (WMMA f32 16x16x4, 25 M-tiles, kept in LDS)
//  phase 4: out[c,px] = sum over 49 dilated taps of x * ker  (VALU, x via L2)
__global__ __launch_bounds__(256) void inv2d_fused(
    const float* __restrict__ x,
    const float* __restrict__ w_reduce,
    const float* __restrict__ b_reduce,
    const float* __restrict__ w_span,
    const float* __restrict__ b_span,
    float* __restrict__ out)
{
  __shared__ float xs[Cn][TPX];     // 8   KB
  __shared__ float red[CrN][TPX];   // 2   KB
  __shared__ float ker[KKG][TPX];   // 24.5 KB  (dynamic kernel stays on-chip)

  const int tid  = threadIdx.x;
  const int lane = tid & 31;
  const int wv   = tid >> 5;

  const int tile = blockIdx.x;          // 0..1023
  const int tw   = tile & 3;            // W / 16 = 4 tiles per row
  const int h    = (tile >> 2) & 63;
  const int b    = tile >> 8;
  const int w0   = tw * TPX;

  // pull the hot shared weight matrices toward the caches (global_prefetch_b8)
  __builtin_prefetch(w_reduce, 0, 0);
  __builtin_prefetch(w_span, 0, 0);

  // ---- phase 1: stage x[:, h, w0:w0+16] into LDS, channel-major ----
  {
    const float* src = x + ((size_t)(b * Cn) * Hn + h) * Wn + w0;
    #pragma unroll
    for (int r = 0; r < 8; ++r) {
      int lin = tid + r * 256;          // 2048 floats, coalesced (px fastest)
      int c   = lin >> 4;
      int px  = lin & 15;
      xs[c][px] = src[(size_t)c * (Hn * Wn) + px];
    }
  }
  __syncthreads();

  const int col = lane & 15;            // N index (pixel)
  const int hi  = lane >> 4;            // lane half: selects K pair / M+8 rows

  // ---- phase 2: red(32x16) = w_reduce(32x128) @ xs(128x16) + b_reduce ----
  if (wv < 2) {
    const int m0 = wv * 16;
    v8f acc = {};
    for (int k0 = 0; k0 < Cn; k0 += 4) {
      const int kk = k0 + 2 * hi;       // A 16x4 layout: lanes0-15 K0/K1, lanes16-31 K2/K3
      v2f a, bf;
      a.x  = w_reduce[(m0 + col) * Cn + kk];
      a.y  = w_reduce[(m0 + col) * Cn + kk + 1];
      bf.x = xs[kk][col];               // B 4x16: row striped across lanes per VGPR
      bf.y = xs[kk + 1][col];
      acc = __builtin_amdgcn_wmma_f32_16x16x4_f32(false, a, false, bf,
                                                  (short)0, acc, false, false);
    }
    #pragma unroll
    for (int r = 0; r < 8; ++r) {       // C/D: VGPR r = rows m0+r (lo half) / m0+r+8 (hi half)
      int row = m0 + r + 8 * hi;
      red[row][col] = acc[r] + b_reduce[row];
    }
  }
  __syncthreads();

  // ---- phase 3: ker(392x16) = w_span(392x32) @ red(32x16) + b_span ----
  for (int t = wv; t < 25; t += 8) {    // 25 M-tiles of 16 (392 padded to 400)
    const int m0 = t * 16;
    const int row_a = m0 + col;
    v8f acc = {};
    for (int k0 = 0; k0 < CrN; k0 += 4) {
      const int kk = k0 + 2 * hi;
      v2f a, bf;
      a.x  = (row_a < KKG) ? w_span[row_a * CrN + kk]     : 0.0f;
      a.y  = (row_a < KKG) ? w_span[row_a * CrN + kk + 1] : 0.0f;
      bf.x = red[kk][col];
      bf.y = red[kk + 1][col];
      acc = __builtin_amdgcn_wmma_f32_16x16x4_f32(false, a, false, bf,
                                                  (short)0, acc, false, false);
    }
    #pragma unroll
    for (int r = 0; r < 8; ++r) {
      int row = m0 + r + 8 * hi;
      if (row < KKG) ker[row][col] = acc[r] + b_span[row];
    }
  }
  __syncthreads();

  // ---- phase 4: dilated 7x7 dynamic-kernel window sum (x served from L2) ----
  {
    const int px = tid & 15;            // pixel within tile
    const int cb = (tid >> 4) * 8;      // 8 channels per thread, all in one group
    const int g  = cb >> 4;             // group id (Cg = 16)
    const int w  = w0 + px;
    float acc[8];
    #pragma unroll
    for (int i = 0; i < 8; ++i) acc[i] = 0.0f;

    const float* xb = x + (size_t)(b * Cn + cb) * (Hn * Wn);
    for (int i = 0; i < Kk; ++i) {
      const int hh = h + DILn * i - PADn;
      if ((unsigned)hh >= (unsigned)Hn) continue;   // zero padding
      for (int j = 0; j < Kk; ++j) {
        const int ww = w + DILn * j - PADn;
        if ((unsigned)ww >= (unsigned)Wn) continue;
        const float kv = ker[g * KKn + i * Kk + j][px];
        const float* xp = xb + hh * Wn + ww;
        #pragma unroll
        for (int cc = 0; cc < 8; ++cc)
          acc[cc] += xp[(size_t)cc * (Hn * Wn)] * kv;
      }
    }

    float* ob = out + ((size_t)(b * Cn + cb) * Hn + h) * Wn + w;
    #pragma unroll
    for (int cc = 0; cc < 8; ++cc)
      ob[(size_t)cc * (Hn * Wn)] = acc[cc];
  }
}

extern "C" void kernel_launch(void* const* d_in, const int* in_sizes, int n_in,
                              void* d_out, int out_size, void* d_ws, size_t ws_size,
                              hipStream_t stream) {
  (void)in_sizes; (void)n_in; (void)out_size; (void)d_ws; (void)ws_size;
  const float* x        = (const float*)d_in[0];
  const float* w_reduce = (const float*)d_in[1];
  const float* b_reduce = (const float*)d_in[2];
  const float* w_span   = (const float*)d_in[3];
  const float* b_span   = (const float*)d_in[4];
  float* out = (float*)d_out;

  const int tiles = Bn * Hn * (Wn / TPX);   // 4 * 64 * 4 = 1024 workgroups
  inv2d_fused<<<dim3(tiles), dim3(256), 0, stream>>>(
      x, w_reduce, b_reduce, w_span, b_span, out);
}